// CVFRLayer_71519795413543
// MI455X (gfx1250) — compile-verified
//
#include <hip/hip_runtime.h>

// Problem constants (match reference)
#define SIZE    2048
#define BATCH   512
#define NCLS    16
#define BLOCKW  128        // SIZE / NCLS
#define STEPS   100
#define DT      0.03f
#define GAMMA   0.125f
#define EIG     1.0f
// sqrt(DT) * EPSILON = sqrt(0.03) * 0.1
#define C_NOISE 0.0173205080756887729f

#define LSTR 36            // LDS row stride (words): mult of 4 -> b128 stores aligned,
                           // 36*L mod 64 distinct for L=0..15 -> conflict-free b64 frags

typedef __attribute__((ext_vector_type(2))) float v2f;
typedef __attribute__((ext_vector_type(8))) float v8f;

#define WMMA_F32(a, b, c) \
  __builtin_amdgcn_wmma_f32_16x16x4_f32(false, (a), false, (b), (short)0, (c), false, false)

// ---------------------------------------------------------------------------
// Setup kernels (once per launch; ~negligible vs 200 step launches)
// ---------------------------------------------------------------------------

// m[n][c] = mean of A[n, c*128 .. c*128+127]
__global__ void rowmean_kernel(const float* __restrict__ A, float* __restrict__ m) {
  int idx = blockIdx.x * blockDim.x + threadIdx.x;      // SIZE*NCLS threads
  if (idx >= SIZE * NCLS) return;
  int n = idx / NCLS, c = idx % NCLS;
  const float* row = A + (size_t)n * SIZE + c * BLOCKW;
  float s = 0.0f;
  for (int i = 0; i < BLOCKW; ++i) s += row[i];
  m[idx] = s * (1.0f / BLOCKW);
}

// Bnk[n][k] = DT * ( A[n][k] - m[n][blk(k)] + (blk(k)==blk(n)) * EIG/128 )
// (this is DT * B^T stored row-major over n -> GEMM B-operand in [n][k] layout;
//  fully coalesced read AND write, no transpose anywhere)
__global__ void build_B_kernel(const float* __restrict__ A,
                               const float* __restrict__ m,
                               float* __restrict__ Bnk) {
  int k = blockIdx.x * blockDim.x + threadIdx.x;        // fast dim (coalesced r+w)
  int n = blockIdx.y;
  float a  = A[(size_t)n * SIZE + k];
  float mm = m[n * NCLS + (k / BLOCKW)];
  float d  = ((k / BLOCKW) == (n / BLOCKW)) ? (EIG * (1.0f / BLOCKW)) : 0.0f;
  Bnk[(size_t)n * SIZE + k] = DT * (a - mm + d);
}

// Gnk = C_NOISE * G  (elementwise scaled copy; [n][k] layout IS G row-major)
__global__ void build_G_kernel(const float* __restrict__ G, float* __restrict__ Gnk) {
  int i = blockIdx.x * blockDim.x + threadIdx.x;
  Gnk[i] = C_NOISE * G[i];
}

// Xnl = x^2 / (GAMMA + x^2), elementwise (float4)
__global__ void nl_kernel(const float4* __restrict__ X, float4* __restrict__ Xnl) {
  int i = blockIdx.x * blockDim.x + threadIdx.x;
  float4 v = X[i];
  float4 o;
  o.x = v.x * v.x; o.y = v.y * v.y; o.z = v.z * v.z; o.w = v.w * v.w;
  o.x = o.x / (GAMMA + o.x); o.y = o.y / (GAMMA + o.y);
  o.z = o.z / (GAMMA + o.z); o.w = o.w / (GAMMA + o.w);
  Xnl[i] = o;
}

// ---------------------------------------------------------------------------
// Fused step GEMM:  Xout = (1-DT)*Xcur + Xnl@B' + Nz@G'
// 64x64 tile / 128-thread block (4 waves); each wave: 32x32 via 2x2 register
// blocking -> 8 f32 WMMAs per 8 fragment loads per k-step.
// ---------------------------------------------------------------------------
__global__ __launch_bounds__(128) void step_kernel(
    const float* __restrict__ Xnl, const float* __restrict__ Nz,
    const float* __restrict__ Bnk, const float* __restrict__ Gnk,
    const float* __restrict__ Xcur, float* __restrict__ Xout) {
  __shared__ float As[64][LSTR];   // Xnl tile    [m][k]
  __shared__ float Ns[64][LSTR];   // noise tile  [m][k]
  __shared__ float Bt[64][LSTR];   // B' tile     [n][k]
  __shared__ float Gt[64][LSTR];   // G' tile     [n][k]

  const int n0 = blockIdx.x * 64;
  const int m0 = blockIdx.y * 64;
  const int t    = threadIdx.x;        // 0..127
  const int lane = t & 31;
  const int w    = t >> 5;             // 0..3
  const int m0w   = (w & 1) * 32;      // wave's 32-row subtile
  const int n0w   = (w >> 1) * 32;     // wave's 32-col subtile
  const int lhalf = lane >> 4;         // selects K pair {0,1} vs {2,3}
  const int l16   = lane & 15;         // M (A-frag) / N (B-frag) index

  v8f acc00 = {}, acc01 = {}, acc10 = {}, acc11 = {};

  for (int kb = 0; kb < SIZE; kb += 32) {
    __syncthreads();
    // Stage all four 64x32 tiles: coalesced b128 global loads -> aligned b128 LDS stores
#pragma unroll
    for (int p = 0; p < 4; ++p) {
      int linear4 = p * 128 + t;          // 0..511 float4 slots per tile
      int r  = linear4 >> 3;              // 0..63
      int c4 = (linear4 & 7) * 4;         // 0,4,...,28
      *(float4*)&As[r][c4] = *(const float4*)&Xnl[(size_t)(m0 + r) * SIZE + kb + c4];
      *(float4*)&Ns[r][c4] = *(const float4*)&Nz [(size_t)(m0 + r) * SIZE + kb + c4];
      *(float4*)&Bt[r][c4] = *(const float4*)&Bnk[(size_t)(n0 + r) * SIZE + kb + c4];
      *(float4*)&Gt[r][c4] = *(const float4*)&Gnk[(size_t)(n0 + r) * SIZE + kb + c4];
    }
    __syncthreads();

#pragma unroll
    for (int kk = 0; kk < 32; kk += 4) {
      const int kb2 = kk + 2 * lhalf;
      // A fragments (16x4): lane = M, lane-halves hold K pairs
      v2f ax0 = *(const v2f*)&As[m0w + l16][kb2];
      v2f ax1 = *(const v2f*)&As[m0w + 16 + l16][kb2];
      v2f an0 = *(const v2f*)&Ns[m0w + l16][kb2];
      v2f an1 = *(const v2f*)&Ns[m0w + 16 + l16][kb2];
      // B fragments (4x16): lane = N, lane-halves hold K pairs
      v2f b0  = *(const v2f*)&Bt[n0w + l16][kb2];
      v2f b1  = *(const v2f*)&Bt[n0w + 16 + l16][kb2];
      v2f g0  = *(const v2f*)&Gt[n0w + l16][kb2];
      v2f g1  = *(const v2f*)&Gt[n0w + 16 + l16][kb2];
      // 2x2 register blocking, both GEMMs into the same accumulators
      acc00 = WMMA_F32(ax0, b0, acc00);
      acc01 = WMMA_F32(ax0, b1, acc01);
      acc10 = WMMA_F32(ax1, b0, acc10);
      acc11 = WMMA_F32(ax1, b1, acc11);
      acc00 = WMMA_F32(an0, g0, acc00);
      acc01 = WMMA_F32(an0, g1, acc01);
      acc10 = WMMA_F32(an1, g0, acc10);
      acc11 = WMMA_F32(an1, g1, acc11);
    }
  }

  // Epilogue: C/D layout -> lane<16: VGPR r = D[M=r, N=lane]; lane>=16: M=8+r
  const float keep = 1.0f - DT;
#pragma unroll
  for (int r = 0; r < 8; ++r) {
    int rowA = m0 + m0w + r + 8 * lhalf;
    int rowB = rowA + 16;
    int col  = n0 + n0w + l16;
    size_t iA = (size_t)rowA * SIZE + col;
    size_t iB = (size_t)rowB * SIZE + col;
    Xout[iA]      = keep * Xcur[iA]      + acc00[r];
    Xout[iA + 16] = keep * Xcur[iA + 16] + acc01[r];
    Xout[iB]      = keep * Xcur[iB]      + acc10[r];
    Xout[iB + 16] = keep * Xcur[iB + 16] + acc11[r];
  }
}

// ---------------------------------------------------------------------------
extern "C" void kernel_launch(void* const* d_in, const int* in_sizes, int n_in,
                              void* d_out, int out_size, void* d_ws, size_t ws_size,
                              hipStream_t stream) {
  const float* x     = (const float*)d_in[0];   // [512, 2048]
  const float* A     = (const float*)d_in[1];   // [2048, 2048]
  const float* G     = (const float*)d_in[2];   // [2048, 2048]
  const float* noise = (const float*)d_in[3];   // [100, 512, 2048]
  float* out = (float*)d_out;

  float* ws   = (float*)d_ws;
  float* Bnk  = ws;                                   // 2048*2048
  float* Gnk  = Bnk + (size_t)SIZE * SIZE;            // 2048*2048
  float* mrow = Gnk + (size_t)SIZE * SIZE;            // 2048*16
  float* X0   = mrow + (size_t)SIZE * NCLS;           // 512*2048
  float* X1   = X0 + (size_t)BATCH * SIZE;            // 512*2048
  float* Xnl  = X1 + (size_t)BATCH * SIZE;            // 512*2048

  // x -> ping buffer (graph-capture safe async d2d copy)
  hipMemcpyAsync(X0, x, sizeof(float) * (size_t)BATCH * SIZE,
                 hipMemcpyDeviceToDevice, stream);

  // Precompute B' (DT * B^T in [n][k]) and G' (scaled G); both live in L2 (33MB)
  rowmean_kernel<<<(SIZE * NCLS + 255) / 256, 256, 0, stream>>>(A, mrow);
  build_B_kernel<<<dim3(SIZE / 256, SIZE), 256, 0, stream>>>(A, mrow, Bnk);
  build_G_kernel<<<(SIZE * SIZE + 255) / 256, 256, 0, stream>>>(G, Gnk);

  float* cur = X0;
  float* nxt = X1;
  for (int s = 0; s < STEPS; ++s) {
    nl_kernel<<<(BATCH * SIZE / 4 + 255) / 256, 256, 0, stream>>>(
        (const float4*)cur, (float4*)Xnl);
    float* dst = (s == STEPS - 1) ? out : nxt;
    step_kernel<<<dim3(SIZE / 64, BATCH / 64), 128, 0, stream>>>(
        Xnl, noise + (size_t)s * BATCH * SIZE, Bnk, Gnk, cur, dst);
    float* tmp = cur; cur = nxt; nxt = tmp;
  }
}